// conv3d_37048387895692
// MI455X (gfx1250) — compile-verified
//
#include <hip/hip_runtime.h>

typedef __attribute__((ext_vector_type(2))) float v2f;
typedef __attribute__((ext_vector_type(8))) float v8f;

#define N_PTS  100000
#define M_PAD  60000
#define C_IN   64
#define C_OUT  64
#define KV     27
#define CENTER 13

// ---------------------------------------------------------------------------
// Center tap: out[m,:] = feats[m,:] @ W[CENTER]  (plain stores, writes all of out)
// One wave -> 16 rows x 64 cols tile. K=64 split into 16 WMMA_F32_16X16X4_F32.
// ---------------------------------------------------------------------------
__global__ __launch_bounds__(256) void spconv_center(const float* __restrict__ feats,
                                                     const float* __restrict__ kern,
                                                     float* __restrict__ out) {
    const int lane  = threadIdx.x & 31;
    const int wave  = threadIdx.x >> 5;
    const int gw    = blockIdx.x * 8 + wave;
    const int nwav  = gridDim.x * 8;
    const int lmod  = lane & 15;   // M (for A) / N (for B,D)
    const int lhi   = lane >> 4;   // 0 or 1: K-pair / row-half selector

    const float* W = kern + CENTER * C_IN * C_OUT;

    // Preload B fragments for the whole 64x64 weight slice.
    // B(4x16) layout (mirror of A): v0 = (K=2*lhi,   N=lmod)
    //                               v1 = (K=2*lhi+1, N=lmod)
    v2f bf[16][4];
#pragma unroll
    for (int kk = 0; kk < 16; ++kk) {
#pragma unroll
        for (int j = 0; j < 4; ++j) {
            const int c = 4 * kk + 2 * lhi;
            bf[kk][j].x = W[(c + 0) * C_OUT + 16 * j + lmod];
            bf[kk][j].y = W[(c + 1) * C_OUT + 16 * j + lmod];
        }
    }

    const int ntiles = N_PTS / 16;  // 6250
    for (int t = gw; t < ntiles; t += nwav) {
        const int m0 = t * 16;
        // A(16x4) f32 layout: lane m (0..15): v0=K0,v1=K1; lane 16+m: v0=K2,v1=K3
        const float* arow = feats + (size_t)(m0 + lmod) * C_IN + 2 * lhi;

        v8f acc[4] = {};
#pragma unroll
        for (int kk = 0; kk < 16; ++kk) {
            v2f a = *(const v2f*)(arow + 4 * kk);
#pragma unroll
            for (int j = 0; j < 4; ++j) {
                acc[j] = __builtin_amdgcn_wmma_f32_16x16x4_f32(
                    false, a, false, bf[kk][j], (short)0, acc[j], false, false);
            }
        }

        // D layout: VGPR i -> (M = i + 8*lhi, N = lmod)
        const int rbase = m0 + 8 * lhi;
#pragma unroll
        for (int j = 0; j < 4; ++j) {
#pragma unroll
            for (int i = 0; i < 8; ++i) {
                out[(size_t)(rbase + i) * C_OUT + 16 * j + lmod] = acc[j][i];
            }
        }
    }
}

// ---------------------------------------------------------------------------
// 26 off-center taps: gather feats rows via imap, GEMM with W[kAct],
// scatter-add to out rows via omap with global f32 atomics.
// blockIdx.y = logical offset k in [0,26); actual weight index skips CENTER.
// ---------------------------------------------------------------------------
__global__ __launch_bounds__(256) void spconv_offsets(const float* __restrict__ feats,
                                                      const float* __restrict__ kern,
                                                      const int* __restrict__ imap,
                                                      const int* __restrict__ omap,
                                                      float* __restrict__ out) {
    const int lane  = threadIdx.x & 31;
    const int wave  = threadIdx.x >> 5;
    const int gw    = blockIdx.x * 8 + wave;
    const int nwav  = gridDim.x * 8;
    const int lmod  = lane & 15;
    const int lhi   = lane >> 4;

    const int k    = blockIdx.y;                 // 0..25
    const int kAct = (k < CENTER) ? k : k + 1;   // skip center slice

    const float* W  = kern + (size_t)kAct * C_IN * C_OUT;
    const int*   im = imap + (size_t)k * M_PAD;
    const int*   om = omap + (size_t)k * M_PAD;

    // Preload B fragments once per wave (amortized over ~8 tiles).
    v2f bf[16][4];
#pragma unroll
    for (int kk = 0; kk < 16; ++kk) {
#pragma unroll
        for (int j = 0; j < 4; ++j) {
            const int c = 4 * kk + 2 * lhi;
            bf[kk][j].x = W[(c + 0) * C_OUT + 16 * j + lmod];
            bf[kk][j].y = W[(c + 1) * C_OUT + 16 * j + lmod];
        }
    }

    const int ntiles = M_PAD / 16;  // 3750
    for (int t = gw; t < ntiles; t += nwav) {
        const int m0 = t * 16;

        // Source row for this lane's A elements (rows are gathered).
        const int  srow  = im[m0 + lmod];
        const bool valid = srow >= 0;
        const float* arow =
            feats + (size_t)(valid ? srow : 0) * C_IN + 2 * lhi;

        v8f acc[4] = {};
#pragma unroll
        for (int kk = 0; kk < 16; ++kk) {
            v2f a = *(const v2f*)(arow + 4 * kk);
            if (!valid) { a.x = 0.0f; a.y = 0.0f; }   // mask invalid rows
#pragma unroll
            for (int j = 0; j < 4; ++j) {
                acc[j] = __builtin_amdgcn_wmma_f32_16x16x4_f32(
                    false, a, false, bf[kk][j], (short)0, acc[j], false, false);
            }
        }

        // Destination rows for this lane's 8 D elements.
        int drow[8];
#pragma unroll
        for (int i = 0; i < 8; ++i) {
            drow[i] = om[m0 + 8 * lhi + i];
        }

#pragma unroll
        for (int i = 0; i < 8; ++i) {
            if (drow[i] >= 0) {
                float* orow = out + (size_t)drow[i] * C_OUT + lmod;
#pragma unroll
                for (int j = 0; j < 4; ++j) {
                    atomicAdd(orow + 16 * j, acc[j][i]);
                }
            }
        }
    }
}

extern "C" void kernel_launch(void* const* d_in, const int* in_sizes, int n_in,
                              void* d_out, int out_size, void* d_ws, size_t ws_size,
                              hipStream_t stream) {
    const float* feats = (const float*)d_in[0];   // (100000, 64) f32
    const float* kern  = (const float*)d_in[1];   // (27, 64, 64) f32
    const int*   imap  = (const int*)d_in[2];     // (26, 60000) i32
    const int*   omap  = (const int*)d_in[3];     // (26, 60000) i32
    float*       out   = (float*)d_out;           // (100000, 64) f32

    // Pass 1: center tap writes the full output tensor (no init needed).
    spconv_center<<<dim3(96), dim3(256), 0, stream>>>(feats, kern, out);

    // Pass 2: 26 gather-GEMM-scatter taps accumulate with f32 atomics.
    spconv_offsets<<<dim3(60, 26), dim3(256), 0, stream>>>(feats, kern, imap, omap, out);
}